// LinearAttention_9216999817220
// MI455X (gfx1250) — compile-verified
//
#include <hip/hip_runtime.h>
#include <math.h>

typedef float v2f __attribute__((ext_vector_type(2)));
typedef float v8f __attribute__((ext_vector_type(8)));

#define LSTRIDE 65   // 64 + 1 pad: kills LDS bank conflicts for column walks
#define SCHUNKS 16   // superchunks per (n,h): 4096 / 16 = 256 rows = 4 chunks of 64

__device__ __forceinline__ v8f wmma4(v2f a, v2f b, v8f c) {
  // V_WMMA_F32_16X16X4_F32 : D = A(16x4,f32) * B(4x16,f32) + C(16x16,f32)
  return __builtin_amdgcn_wmma_f32_16x16x4_f32(
      /*neg_a=*/false, a, /*neg_b=*/false, b,
      /*c_mod=*/(short)0, c, /*reuse_a=*/false, /*reuse_b=*/false);
}

// A fragment: 16x4 tile at [row0, col0] of row-major smem.
__device__ __forceinline__ v2f afrag(const float* s, int row0, int col0, int lane) {
  int r = row0 + (lane & 15);
  int c = col0 + ((lane >> 4) << 1);
  const float* p = s + r * LSTRIDE + c;
  v2f a; a.x = p[0]; a.y = p[1];
  return a;
}

// B fragment: 4x16 tile at [row0, col0] of row-major smem.
__device__ __forceinline__ v2f bfrag(const float* s, int row0, int col0, int lane) {
  int c = col0 + (lane & 15);
  int r = row0 + ((lane >> 4) << 1);
  const float* p = s + r * LSTRIDE + c;
  v2f b; b.x = p[0]; b.y = p[LSTRIDE];
  return b;
}

// A fragment of the *transpose* of smem: A[i][j] = s[j][i]
__device__ __forceinline__ v2f afragT(const float* s, int row0, int col0, int lane) {
  int r = row0 + (lane & 15);
  int c = col0 + ((lane >> 4) << 1);
  const float* p = s + c * LSTRIDE + r;
  v2f a; a.x = p[0]; a.y = p[LSTRIDE];
  return a;
}

__device__ __forceinline__ float elu1(float x) {
  return x > 0.f ? x + 1.f : __expf(x);   // elu(x)+1
}

// ============================================================================
// Pass 1: per-superchunk partial state  dS = sum kf^T v ,  dkvec = colsum(kf)
// grid = N*H*SCHUNKS, block = 512 (16 wave32s tiling the 64x64 state)
// ============================================================================
__global__ __launch_bounds__(512)
void linattn_pass1_partials(const float* __restrict__ k,
                            const float* __restrict__ v,
                            float* __restrict__ wsS,   // [N*H, SC, 64*64]
                            float* __restrict__ wsK) { // [N*H, SC, 64]
  constexpr int L = 4096, H = 16, DIM = 64;
  constexpr int SROWS = L / SCHUNKS;          // 256 rows per superchunk
  __shared__ float sK[64 * LSTRIDE], sV[64 * LSTRIDE];

  const int b    = blockIdx.x;
  const int nh   = b / SCHUNKS;
  const int sc   = b % SCHUNKS;
  const int n    = nh / H, h = nh % H;
  const int tid  = threadIdx.x;
  const int lane = tid & 31;
  const int wv   = tid >> 5;
  const int mi   = wv >> 2, ni = wv & 3;
  const int crow = (lane >> 4) << 3;
  const int ccol = lane & 15;

  const int lrow = tid >> 3;
  const int lcol = (tid & 7) * 8;
  const size_t rowstride = (size_t)H * DIM;

  v8f acc = {};
  float kv = 0.f;

  for (int c = 0; c < SROWS / 64; ++c) {
    const int l0 = sc * SROWS + c * 64;
    {
      size_t g = ((size_t)n * L + (size_t)(l0 + lrow)) * rowstride
               + (size_t)h * DIM + lcol;
      const float4 ka = *(const float4*)(k + g);
      const float4 kb = *(const float4*)(k + g + 4);
      const float4 va = *(const float4*)(v + g);
      const float4 vb = *(const float4*)(v + g + 4);
      float* dk = sK + lrow * LSTRIDE + lcol;
      float* dv = sV + lrow * LSTRIDE + lcol;
      dk[0] = elu1(ka.x); dk[1] = elu1(ka.y); dk[2] = elu1(ka.z); dk[3] = elu1(ka.w);
      dk[4] = elu1(kb.x); dk[5] = elu1(kb.y); dk[6] = elu1(kb.z); dk[7] = elu1(kb.w);
      dv[0] = va.x; dv[1] = va.y; dv[2] = va.z; dv[3] = va.w;
      dv[4] = vb.x; dv[5] = vb.y; dv[6] = vb.z; dv[7] = vb.w;
    }
    __syncthreads();

    if (tid < 64) {                   // waves 0,1 fully -> wave-uniform
      for (int j = 0; j < 64; ++j) kv += sK[j * LSTRIDE + tid];
    }
#pragma unroll
    for (int kk = 0; kk < 16; ++kk) {
      v2f a = afragT(sK, mi * 16, kk * 4, lane);   // (Kf^T)[d][j]
      v2f bb = bfrag(sV, kk * 4, ni * 16, lane);
      acc = wmma4(a, bb, acc);
    }
    __syncthreads();                  // before next chunk overwrites LDS
  }

  const size_t base = ((size_t)nh * SCHUNKS + sc) * 4096;
#pragma unroll
  for (int r = 0; r < 8; ++r) {
    int row = mi * 16 + r + crow;
    int col = ni * 16 + ccol;
    wsS[base + row * 64 + col] = acc[r];
  }
  if (tid < 64) wsK[((size_t)nh * SCHUNKS + sc) * 64 + tid] = kv;
}

// ============================================================================
// Pass 2: in-place EXCLUSIVE prefix over superchunks (per (n,h))
// grid = N*H, block = 256
// ============================================================================
__global__ __launch_bounds__(256)
void linattn_pass2_prefix(float* __restrict__ wsS, float* __restrict__ wsK) {
  const int nh  = blockIdx.x;
  const int tid = threadIdx.x;

  float run[16];
#pragma unroll
  for (int i = 0; i < 16; ++i) run[i] = 0.f;
  for (int sc = 0; sc < SCHUNKS; ++sc) {
    float* p = wsS + ((size_t)nh * SCHUNKS + sc) * 4096 + tid * 16;
#pragma unroll
    for (int i = 0; i < 16; ++i) {
      float t = p[i];
      p[i] = run[i];
      run[i] += t;
    }
  }
  if (tid < 64) {
    float rk = 0.f;
    for (int sc = 0; sc < SCHUNKS; ++sc) {
      size_t idx = ((size_t)nh * SCHUNKS + sc) * 64 + tid;
      float t = wsK[idx];
      wsK[idx] = rk;
      rk += t;
    }
  }
}

// ============================================================================
// Pass 3: chunked causal linear attention with seeded state.
// grid = N*H*SCHUNKS (or N*H in fallback mode), block = 512.
// seeded==0 -> fallback single-pass over all of L with zero initial state.
// ============================================================================
__global__ __launch_bounds__(512)
void linattn_pass3_chunked(const float* __restrict__ q,
                           const float* __restrict__ k,
                           const float* __restrict__ v,
                           float* __restrict__ out,
                           const float* __restrict__ wsS,
                           const float* __restrict__ wsK,
                           int seeded) {
  constexpr int L = 4096, H = 16, DIM = 64, CH = 64;
  __shared__ float sQ[64 * LSTRIDE], sK[64 * LSTRIDE], sV[64 * LSTRIDE];
  __shared__ float sP[64 * LSTRIDE];   // masked intra-chunk scores
  __shared__ float sS[64 * LSTRIDE];   // running state  S = sum k v^T
  __shared__ float sZ[64];             // per-row normalizer
  __shared__ float sVec[64];           // running cumsum of k features

  const int b    = blockIdx.x;
  const int nh   = seeded ? (b / SCHUNKS) : b;
  const int sc   = seeded ? (b % SCHUNKS) : 0;
  const int n    = nh / H, h = nh % H;
  const int tid  = threadIdx.x;
  const int lane = tid & 31;
  const int wv   = tid >> 5;
  const int mi   = wv >> 2, ni = wv & 3;
  const int crow = (lane >> 4) << 3;
  const int ccol = lane & 15;

  // ---- seed state from exclusive-prefix workspace (or zero) ----
  if (seeded) {
    const size_t base = ((size_t)nh * SCHUNKS + sc) * 4096;
#pragma unroll
    for (int r = 0; r < 8; ++r) {
      int row = mi * 16 + r + crow;
      int col = ni * 16 + ccol;
      sS[row * LSTRIDE + col] = wsS[base + row * 64 + col];
    }
    if (tid < 64) sVec[tid] = wsK[((size_t)nh * SCHUNKS + sc) * 64 + tid];
  } else {
#pragma unroll
    for (int r = 0; r < 8; ++r)
      sS[(mi * 16 + r + crow) * LSTRIDE + ni * 16 + ccol] = 0.f;
    if (tid < 64) sVec[tid] = 0.f;
  }
  __syncthreads();

  const int lrow = tid >> 3;
  const int lcol = (tid & 7) * 8;
  const size_t rowstride = (size_t)H * DIM;

  const int nchunks = seeded ? (L / SCHUNKS / CH) : (L / CH);
  const int lbase   = sc * (L / SCHUNKS);

  for (int ch = 0; ch < nchunks; ++ch) {
    const int l0 = lbase + ch * CH;

    // ---------- load chunk, apply elu+1 feature map to q,k ----------
    {
      size_t g = ((size_t)n * L + (size_t)(l0 + lrow)) * rowstride
               + (size_t)h * DIM + lcol;
      const float4 qa = *(const float4*)(q + g);
      const float4 qb = *(const float4*)(q + g + 4);
      const float4 ka = *(const float4*)(k + g);
      const float4 kb = *(const float4*)(k + g + 4);
      const float4 va = *(const float4*)(v + g);
      const float4 vb = *(const float4*)(v + g + 4);
      float* dq = sQ + lrow * LSTRIDE + lcol;
      float* dk = sK + lrow * LSTRIDE + lcol;
      float* dv = sV + lrow * LSTRIDE + lcol;
      dq[0] = elu1(qa.x); dq[1] = elu1(qa.y); dq[2] = elu1(qa.z); dq[3] = elu1(qa.w);
      dq[4] = elu1(qb.x); dq[5] = elu1(qb.y); dq[6] = elu1(qb.z); dq[7] = elu1(qb.w);
      dk[0] = elu1(ka.x); dk[1] = elu1(ka.y); dk[2] = elu1(ka.z); dk[3] = elu1(ka.w);
      dk[4] = elu1(kb.x); dk[5] = elu1(kb.y); dk[6] = elu1(kb.z); dk[7] = elu1(kb.w);
      dv[0] = va.x; dv[1] = va.y; dv[2] = va.z; dv[3] = va.w;
      dv[4] = vb.x; dv[5] = vb.y; dv[6] = vb.z; dv[7] = vb.w;
    }
    __syncthreads();

    // ---------- intra-chunk scores  P = causal_mask(Qf * Kf^T) ----------
    if (ni <= mi) {
      v8f acc = {};
#pragma unroll
      for (int kk = 0; kk < 16; ++kk) {
        v2f a = afrag(sQ, mi * 16, kk * 4, lane);
        v2f bb = afrag(sK, ni * 16, kk * 4, lane);  // B[d][j] = Kf[j][d]
        acc = wmma4(a, bb, acc);
      }
#pragma unroll
      for (int r = 0; r < 8; ++r) {
        int row = mi * 16 + r + crow;
        int col = ni * 16 + ccol;
        sP[row * LSTRIDE + col] = (col <= row) ? acc[r] : 0.f;
      }
    }
    __syncthreads();

    // ---------- normalizer  z_i = eps + q_i . s_prev + rowsum(P_i) ----------
    if (tid < 64) {
      float zi = 1e-6f;
      const float* qr = sQ + tid * LSTRIDE;
      for (int d2 = 0; d2 < 64; ++d2) zi += qr[d2] * sVec[d2];
      const float* pr = sP + tid * LSTRIDE;
      for (int j = 0; j <= tid; ++j) zi += pr[j];
      sZ[tid] = zi;
    }

    // ---------- O = P*V  +  Qf*S_prev ----------
    v8f o = {};
    for (int kb = 0; kb <= mi; ++kb) {
#pragma unroll
      for (int ks = 0; ks < 4; ++ks) {
        v2f a = afrag(sP, mi * 16, kb * 16 + ks * 4, lane);
        v2f bb = bfrag(sV, kb * 16 + ks * 4, ni * 16, lane);
        o = wmma4(a, bb, o);
      }
    }
#pragma unroll
    for (int kk = 0; kk < 16; ++kk) {
      v2f a = afrag(sQ, mi * 16, kk * 4, lane);
      v2f bb = bfrag(sS, kk * 4, ni * 16, lane);
      o = wmma4(a, bb, o);
    }
    __syncthreads();   // all reads of sS / writes of sZ complete

    // ---------- normalize + store output ----------
#pragma unroll
    for (int r = 0; r < 8; ++r) {
      int row = mi * 16 + r + crow;
      int col = ni * 16 + ccol;
      size_t g = ((size_t)n * L + (size_t)(l0 + row)) * rowstride
               + (size_t)h * DIM + col;
      out[g] = o[r] / sZ[row];
    }

    // ---------- state update:  S += Kf^T * V ,  sVec += colsum(Kf) ----------
    v8f dS = {};
#pragma unroll
    for (int kk = 0; kk < 16; ++kk) {
      v2f a = afragT(sK, mi * 16, kk * 4, lane);
      v2f bb = bfrag(sV, kk * 4, ni * 16, lane);
      dS = wmma4(a, bb, dS);
    }
#pragma unroll
    for (int r = 0; r < 8; ++r) {
      int row = mi * 16 + r + crow;
      int col = ni * 16 + ccol;
      sS[row * LSTRIDE + col] += dS[r];
    }
    if (tid < 64) {
      float s = 0.f;
      for (int j = 0; j < 64; ++j) s += sK[j * LSTRIDE + tid];
      sVec[tid] += s;
    }
    __syncthreads();
  }
}

extern "C" void kernel_launch(void* const* d_in, const int* in_sizes, int n_in,
                              void* d_out, int out_size, void* d_ws, size_t ws_size,
                              hipStream_t stream) {
  (void)n_in; (void)out_size;
  const float* q = (const float*)d_in[0];
  const float* k = (const float*)d_in[1];
  const float* v = (const float*)d_in[2];
  float* out = (float*)d_out;
  const int L = 4096, H = 16, D = 64;
  const int N = in_sizes[0] / (L * H * D);   // = 4
  const int NH = N * H;

  const size_t needS = (size_t)NH * SCHUNKS * 4096;   // floats
  const size_t needK = (size_t)NH * SCHUNKS * 64;     // floats
  const size_t needBytes = (needS + needK) * sizeof(float);

  if (d_ws != nullptr && ws_size >= needBytes) {
    float* wsS = (float*)d_ws;
    float* wsK = wsS + needS;
    // Pass 1: per-superchunk partial states (fully parallel WMMA)
    linattn_pass1_partials<<<dim3(NH * SCHUNKS), dim3(512), 0, stream>>>(k, v, wsS, wsK);
    // Pass 2: exclusive prefix over superchunks
    linattn_pass2_prefix<<<dim3(NH), dim3(256), 0, stream>>>(wsS, wsK);
    // Pass 3: seeded chunked attention, 16x the parallelism
    linattn_pass3_chunked<<<dim3(NH * SCHUNKS), dim3(512), 0, stream>>>(
        q, k, v, out, wsS, wsK, /*seeded=*/1);
  } else {
    // Fallback: single-pass over the whole sequence per (n,h)
    linattn_pass3_chunked<<<dim3(NH), dim3(512), 0, stream>>>(
        q, k, v, out, nullptr, nullptr, /*seeded=*/0);
  }
}